// Renderer_51273319580150
// MI455X (gfx1250) — compile-verified
//
#include <hip/hip_runtime.h>
#include <math.h>

// MI455X (gfx1250) fused neural-stroke renderer.
//  - Brushes (2 MB) are L2-resident (192 MB L2) -> gather/VALU-bound, not HBM-bound.
//  - Fully fused: no 128 MB intermediate stroke/alpha images; canvas kept in VGPRs.
//  - Sequential over-composite split into S chunks (affine composition is
//    associative): per chunk keep T = prod(1-a), B = over-accumulated color.
//  - Grid einsum ('hwk,nok->nhwo', K=3) mapped onto V_WMMA_F32_16X16X4_F32:
//    one WMMA gives gx (or gy) for a whole 16x16 pixel tile.
//  - wave32: 1 wave per block, each lane owns 1 row x 8 cols of the tile
//    (matches the f32 16x16 WMMA D layout exactly).
//  - Per-stroke warp/color/brush-select hoisted into a prepass kernel so the
//    hot loop has no transcendentals / IEEE-div expansions (asm showed
//    v_div_fmas/v_div_fixup + ocml sin/cos recomputed 64x per stroke).

typedef __attribute__((ext_vector_type(2))) float v2f;
typedef __attribute__((ext_vector_type(8))) float v8f;

#define IMG 128
#define BR  512
#define SCREC 12   // floats per stroke-constant record

__device__ __forceinline__ float brush_tap(const float* __restrict__ img, int ix, int iy) {
  bool valid = (ix >= 0) & (ix < BR) & (iy >= 0) & (iy < BR);
  int cx = min(max(ix, 0), BR - 1);
  int cy = min(max(iy, 0), BR - 1);
  float v = img[cy * BR + cx];
  return valid ? v : 0.0f;
}

// reference _bilinear_sample, Wi=Hi=512: x = ((gx+1)*512 - 1)*0.5
__device__ __forceinline__ float bilin(const float* __restrict__ img, float gx, float gy) {
  float x = (gx + 1.0f) * 256.0f - 0.5f;
  float y = (gy + 1.0f) * 256.0f - 0.5f;
  float xf = floorf(x), yf = floorf(y);
  int x0 = (int)xf, y0 = (int)yf;
  float wx1 = x - xf, wy1 = y - yf;
  float wx0 = 1.0f - wx1, wy0 = 1.0f - wy1;
  float t00 = brush_tap(img, x0,     y0);
  float t10 = brush_tap(img, x0 + 1, y0);
  float t01 = brush_tap(img, x0,     y0 + 1);
  float t11 = brush_tap(img, x0 + 1, y0 + 1);
  return t00 * (wx0 * wy0) + t10 * (wx1 * wy0) + t01 * (wx0 * wy1) + t11 * (wx1 * wy1);
}

// reference _nearest_sample of alpha_src = (brush > 0); jnp.round = rint (half-even)
__device__ __forceinline__ float nearest_alpha(const float* __restrict__ img, float gx, float gy) {
  float x = (gx + 1.0f) * 256.0f - 0.5f;
  float y = (gy + 1.0f) * 256.0f - 0.5f;
  int ix = (int)rintf(x);
  int iy = (int)rintf(y);
  bool valid = (ix >= 0) & (ix < BR) & (iy >= 0) & (iy < BR);
  int cx = min(max(ix, 0), BR - 1);
  int cy = min(max(iy, 0), BR - 1);
  float v = img[cy * BR + cx];
  return (valid && (v > 0.0f)) ? 1.0f : 0.0f;
}

// Compute per-stroke constants exactly as the reference expresses them
// (same sinf/cosf, same exact divisions, same FMA-able combinations).
__device__ __forceinline__ void stroke_warp(const float* __restrict__ pp, float* sc10) {
  float x0 = pp[0], y0 = pp[1], w = pp[2], h = pp[3], th = pp[4];
  float sin_t = sinf(3.14159265358979323846f * th);
  float cos_t = cosf(3.14159265358979323846f * th);
  // H == W == 128, so H/W factors cancel.
  float w00 = cos_t / w;
  float w01 = sin_t / w;
  float w02 = (1.0f - 2.0f * x0) * w00 + (1.0f - 2.0f * y0) * w01;
  float w10 = -sin_t / h;
  float w11 = cos_t / h;
  float w12 = (1.0f - 2.0f * y0) * w11 + (1.0f - 2.0f * x0) * w10;
  sc10[0] = w00; sc10[1] = w01; sc10[2] = w02;
  sc10[3] = w10; sc10[4] = w11; sc10[5] = w12;
  sc10[6] = pp[5]; sc10[7] = pp[6]; sc10[8] = pp[7];
  sc10[9] = (h > w) ? 0.0f : 1.0f;   // brush select: idx = where(h>w, 0, 1)
}

// Prepass: one thread per stroke (N = 1024), hoists transcendentals + divisions
// out of the 64-blocks-per-chunk hot loop.
__global__ __launch_bounds__(256)
void stroke_constants(const float* __restrict__ params, float* __restrict__ sc) {
  int n = blockIdx.x * 256 + threadIdx.x;
  if (n >= 1024) return;
  float rec[10];
  stroke_warp(params + (size_t)n * 8, rec);
  float* dst = sc + (size_t)n * SCREC;
#pragma unroll
  for (int i = 0; i < 10; ++i) dst[i] = rec[i];
  dst[10] = 0.0f; dst[11] = 0.0f;
}

// One wave (32 threads) per 16x16 pixel tile per (batch, stroke-chunk).
// grid = (8, 8, 4*S). z = k*4 + b. Each chunk covers strokes [k*CH, (k+1)*CH).
// Outputs per chunk: outB[z][3][128*128] (over-accum color), outT[z][128*128]
// (transmittance). For S==1 pass outT = nullptr and outB = d_out (F(0) = B).
__global__ __launch_bounds__(32)
void render_chunks(const float* __restrict__ params,   // (4,256,8)
                   const float* __restrict__ brushes,  // (2,1,512,512)
                   const float* __restrict__ sc,       // stroke constants or null
                   float* __restrict__ outB,
                   float* __restrict__ outT,
                   int chunkLen)
{
  __shared__ float atile[18 * 20];   // 18x18 alpha halo tile, stride 20 (bank spread)

  const int lane = threadIdx.x;               // 0..31
  const int PX0  = blockIdx.x * 16;           // tile origin (x = col)
  const int PY0  = blockIdx.y * 16;           // tile origin (y = row)
  const int z    = blockIdx.z;
  const int b    = z & 3;
  const int k    = z >> 2;
  const int row  = lane & 15;                 // my pixel row within tile (WMMA N)
  const int colh = (lane >> 4) * 8;           // my 8-col half (WMMA M = colh + r)

  // --- WMMA A matrix (stroke-invariant): A[m][0]=xs_m, A[m][1]=1, K=2,3 zero.
  // 16x4 f32 A layout: lanes0-15 hold K=0/1 in VGPR0/1 (M=lane); lanes16-31 hold K=2/3.
  float bx_l = (2.0f * (float)(PX0 + lane) + 1.0f) * (1.0f / 128.0f) - 1.0f;
  v2f A;
  A.x = (lane < 16) ? bx_l : 0.0f;
  A.y = (lane < 16) ? 1.0f : 0.0f;
  // base-y coordinate for column n=lane of B (rows of the tile)
  float by_l = (2.0f * (float)(PY0 + row) + 1.0f) * (1.0f / 128.0f) - 1.0f;

  // Per-pixel affine-composite state (8 pixels per lane), canvas stays in VGPRs.
  float T[8];
  float Bc[3][8];
#pragma unroll
  for (int r = 0; r < 8; ++r) {
    T[r] = 1.0f;
    Bc[0][r] = 0.0f; Bc[1][r] = 0.0f; Bc[2][r] = 0.0f;
  }

  for (int li = 0; li < chunkLen; ++li) {
    const int l = k * chunkLen + li;
    const int n = b * 256 + l;

    float w00, w01, w02, w10, w11, w12, cR, cG, cBv, sel;
    if (sc) {            // uniform branch; constants via scalar loads from L2
      const float* cp = sc + (size_t)n * SCREC;
      w00 = cp[0]; w01 = cp[1]; w02 = cp[2];
      w10 = cp[3]; w11 = cp[4]; w12 = cp[5];
      cR = cp[6]; cG = cp[7]; cBv = cp[8]; sel = cp[9];
    } else {             // fallback: tiny-workspace case, compute inline
      float rec[10];
      stroke_warp(params + (size_t)n * 8, rec);
      w00 = rec[0]; w01 = rec[1]; w02 = rec[2];
      w10 = rec[3]; w11 = rec[4]; w12 = rec[5];
      cR = rec[6]; cG = rec[7]; cBv = rec[8]; sel = rec[9];
    }
    const float* __restrict__ img = brushes + ((sel != 0.0f) ? (BR * BR) : 0);

    // --- grid einsum via matrix core:
    // gx[m][n] = xs_m*w00 + (w01*ys_n + w02) ; gy[m][n] = xs_m*w10 + (w11*ys_n + w12)
    v2f Bx, By;
    Bx.x = (lane < 16) ? w00 : 0.0f;
    Bx.y = (lane < 16) ? (w01 * by_l + w02) : 0.0f;
    By.x = (lane < 16) ? w10 : 0.0f;
    By.y = (lane < 16) ? (w11 * by_l + w12) : 0.0f;
    v8f cz = {};
    v8f gx = __builtin_amdgcn_wmma_f32_16x16x4_f32(false, A, false, Bx, (short)0, cz, false, false);
    v8f gy = __builtin_amdgcn_wmma_f32_16x16x4_f32(false, A, false, By, (short)0, cz, false, false);

    // --- alpha on 18x18 halo tile (reference pads with 1e9 outside the image)
    for (int i = lane; i < 18 * 18; i += 32) {
      int hr = i / 18, hc = i - hr * 18;
      int py = PY0 + hr - 1, px = PX0 + hc - 1;
      float a;
      if ((unsigned)px >= 128u || (unsigned)py >= 128u) {
        a = 1.0e9f;
      } else {
        float bx = (2.0f * (float)px + 1.0f) * (1.0f / 128.0f) - 1.0f;
        float by = (2.0f * (float)py + 1.0f) * (1.0f / 128.0f) - 1.0f;
        a = nearest_alpha(img, w00 * bx + w01 * by + w02,
                               w10 * bx + w11 * by + w12);
      }
      atile[hr * 20 + hc] = a;
    }
    __syncthreads();

    // --- separable 3x3 min erosion: vertical 3-min over my 3 rows, 10 cols
    float vm[10];
#pragma unroll
    for (int c = 0; c < 10; ++c) {
      float a0 = atile[(row + 0) * 20 + colh + c];
      float a1 = atile[(row + 1) * 20 + colh + c];
      float a2 = atile[(row + 2) * 20 + colh + c];
      vm[c] = fminf(a0, fminf(a1, a2));
    }

    // --- bilinear sample + tint + affine-composite update for my 8 pixels
#pragma unroll
    for (int r = 0; r < 8; ++r) {
      float a  = fminf(vm[r], fminf(vm[r + 1], vm[r + 2]));   // eroded alpha
      float s  = bilin(img, gx[r], gy[r]);
      // c_new = fg*a + c*(1-a)  ==  c + a*(fg - c);  T_new = T*(1-a)
      Bc[0][r] = fmaf(a, fmaf(s, cR,  -Bc[0][r]), Bc[0][r]);
      Bc[1][r] = fmaf(a, fmaf(s, cG,  -Bc[1][r]), Bc[1][r]);
      Bc[2][r] = fmaf(a, fmaf(s, cBv, -Bc[2][r]), Bc[2][r]);
      T[r]     = fmaf(-a, T[r], T[r]);
    }
    __syncthreads();   // atile reused next stroke
  }

  // --- store: 8 consecutive floats per channel per lane -> 2x float4, aligned.
  const int pixBase = (PY0 + row) * 128 + PX0 + colh;
#pragma unroll
  for (int ch = 0; ch < 3; ++ch) {
    float4* dst = (float4*)(outB + ((size_t)(z * 3 + ch) << 14) + pixBase);
    dst[0] = make_float4(Bc[ch][0], Bc[ch][1], Bc[ch][2], Bc[ch][3]);
    dst[1] = make_float4(Bc[ch][4], Bc[ch][5], Bc[ch][6], Bc[ch][7]);
  }
  if (outT) {
    float4* dst = (float4*)(outT + ((size_t)z << 14) + pixBase);
    dst[0] = make_float4(T[0], T[1], T[2], T[3]);
    dst[1] = make_float4(T[4], T[5], T[6], T[7]);
  }
}

// Fold the S chunk-affine maps in stroke order: c = B_k + T_k * c, k = 0..S-1.
__global__ __launch_bounds__(256)
void combine_chunks(const float* __restrict__ wsB, const float* __restrict__ wsT,
                    float* __restrict__ out, int S)
{
  int t = blockIdx.x * 256 + threadIdx.x;   // 0 .. 4*16384-1
  int b = t >> 14;
  int pix = t & 16383;
  float c0 = 0.0f, c1 = 0.0f, c2 = 0.0f;
  for (int kk = 0; kk < S; ++kk) {
    int z = kk * 4 + b;
    float Tk = wsT[((size_t)z << 14) + pix];
    const float* Bp = wsB + ((size_t)(z * 3) << 14) + pix;
    c0 = fmaf(Tk, c0, Bp[0]);
    c1 = fmaf(Tk, c1, Bp[1 << 14]);
    c2 = fmaf(Tk, c2, Bp[2 << 14]);
  }
  out[((size_t)(b * 3 + 0) << 14) + pix] = c0;
  out[((size_t)(b * 3 + 1) << 14) + pix] = c1;
  out[((size_t)(b * 3 + 2) << 14) + pix] = c2;
}

extern "C" void kernel_launch(void* const* d_in, const int* in_sizes, int n_in,
                              void* d_out, int out_size, void* d_ws, size_t ws_size,
                              hipStream_t stream) {
  (void)in_sizes; (void)n_in; (void)out_size;
  const float* params  = (const float*)d_in[0];   // (4,256,8) f32
  const float* brushes = (const float*)d_in[1];   // (2,1,512,512) f32
  float* out = (float*)d_out;                     // (4,3,128,128) f32

  const size_t HW = 128 * 128;
  const size_t scBytes = (size_t)1024 * SCREC * sizeof(float);   // 48 KB

  // Pick chunk count S (parallelism over the sequential scan) to fit workspace:
  // per chunk-batch-pixel we store 3 (B) + 1 (T) floats, plus the 48 KB table.
  int S = 8;
  while (S > 1 && (size_t)S * 4 * 4 * HW * sizeof(float) + scBytes > ws_size) S >>= 1;

  const bool haveSC = ((size_t)S * 4 * 4 * HW * sizeof(float) * (S > 1 ? 1 : 0)
                       + scBytes) <= ws_size;
  float* wsB = (float*)d_ws;
  float* wsT = wsB + (size_t)S * 4 * 3 * HW;
  float* scTab = nullptr;
  if (haveSC) {
    scTab = (S > 1) ? (wsT + (size_t)S * 4 * HW) : (float*)d_ws;
    stroke_constants<<<4, 256, 0, stream>>>(params, scTab);
  }

  dim3 grid(8, 8, 4 * S);
  if (S == 1) {
    // c_final = F(0) = B, and B-layout [b][3][HW] == d_out layout.
    render_chunks<<<grid, 32, 0, stream>>>(params, brushes, scTab, out, nullptr, 256);
  } else {
    render_chunks<<<grid, 32, 0, stream>>>(params, brushes, scTab, wsB, wsT, 256 / S);
    combine_chunks<<<256, 256, 0, stream>>>(wsB, wsT, out, S);
  }
}